// HGATLayer_26680336843177
// MI455X (gfx1250) — compile-verified
//
#include <hip/hip_runtime.h>
#include <math.h>

#define CURV     0.01f
#define SQRTC    0.1f
#define EPSV     1e-6f
#define MIN_NRM  1e-10f
#define ENC_NEG_INF 0x007FFFFFu

typedef __attribute__((ext_vector_type(2))) float v2f;
typedef __attribute__((ext_vector_type(8))) float v8f;

// ---------- wave32 helpers ----------
__device__ __forceinline__ float wave_reduce_add(float v) {
#pragma unroll
    for (int m = 16; m >= 1; m >>= 1) v += __shfl_xor(v, m, 32);
    return v;
}
__device__ __forceinline__ float half_reduce_add(float v) { // within 16-lane half
#pragma unroll
    for (int m = 8; m >= 1; m >>= 1) v += __shfl_xor(v, m, 32);
    return v;
}
__device__ __forceinline__ int wave_reduce_min(int v) {
#pragma unroll
    for (int m = 16; m >= 1; m >>= 1) { int o = __shfl_xor(v, m, 32); v = o < v ? o : v; }
    return v;
}
__device__ __forceinline__ unsigned enc_f(float f) {
    unsigned u = __float_as_uint(f);
    return (u & 0x80000000u) ? ~u : (u | 0x80000000u);
}
__device__ __forceinline__ float dec_f(unsigned u) {
    unsigned b = (u & 0x80000000u) ? (u ^ 0x80000000u) : ~u;
    return __uint_as_float(b);
}

// ---------- K0: init workspace ----------
__global__ void k_init(unsigned* m_enc, float* denom, float* num, float* den,
                       unsigned* cnt, int* sorted, int N, int Lsorted) {
    int stride = gridDim.x * blockDim.x;
    int i0 = blockIdx.x * blockDim.x + threadIdx.x;
    for (int i = i0; i < N * 4; i += stride) { m_enc[i] = ENC_NEG_INF; denom[i] = 0.f; den[i] = 0.f; }
    for (int i = i0; i < N * 256; i += stride) num[i] = 0.f;
    for (int i = i0; i < 8; i += stride) cnt[i] = 0u;
    for (int i = i0; i < Lsorted; i += stride) sorted[i] = -1;
}

// ---------- K1: tangent features (log_map_zero), wave per node ----------
__global__ void k_logmap(const float* __restrict__ h, float* __restrict__ ht, int N) {
    int wave = threadIdx.x >> 5, lane = threadIdx.x & 31;
    int node = blockIdx.x * (blockDim.x >> 5) + wave;
    if (node >= N) return;
    v2f x = ((const v2f*)(h + (size_t)node * 64))[lane];
    float nrm = fmaxf(sqrtf(wave_reduce_add(x.x * x.x + x.y * x.y)), MIN_NRM);
    float t = fminf(fmaxf(SQRTC * nrm, MIN_NRM), 1.0f - 1e-5f);
    float s = atanhf(t) / t;
    v2f o; o.x = s * x.x; o.y = s * x.y;
    ((v2f*)(ht + (size_t)node * 64))[lane] = o;
}

// ---------- counting sort of edges by relation ----------
__global__ void k_hist(const int* __restrict__ et, unsigned* cnt, int E) {
    int e = blockIdx.x * blockDim.x + threadIdx.x;
    if (e < E) atomicAdd(&cnt[et[e]], 1u);
}
__global__ void k_offsets(const unsigned* __restrict__ cnt, unsigned* cursor) {
    unsigned o = 0;
    for (int r = 0; r < 8; ++r) { cursor[r] = o; o += (cnt[r] + 15u) & ~15u; }
}
__global__ void k_scatter(const int* __restrict__ et, unsigned* cursor, int* sorted, int E) {
    int e = blockIdx.x * blockDim.x + threadIdx.x;
    if (e < E) { unsigned p = atomicAdd(&cursor[et[e]], 1u); sorted[p] = e; }
}

// ---------- K2: attention score + segment max, wave per edge ----------
__global__ void k_score(const float* __restrict__ h, const float* __restrict__ av,
                        const int* __restrict__ src, const int* __restrict__ dst,
                        const int* __restrict__ et,
                        float* __restrict__ score, unsigned* __restrict__ m_enc, int E) {
    int wave = threadIdx.x >> 5, lane = threadIdx.x & 31;
    int e = blockIdx.x * (blockDim.x >> 5) + wave;
    if (e >= E) return;
    int s = src[e], d = dst[e], r = et[e];
    v2f xv = ((const v2f*)(h + (size_t)s * 64))[lane];
    v2f yv = ((const v2f*)(h + (size_t)d * 64))[lane];
    float x0 = xv.x, x1 = xv.y;
    float y0 = -yv.x, y1 = -yv.y;
    float xy = wave_reduce_add(x0 * y0 + x1 * y1);
    float x2 = wave_reduce_add(x0 * x0 + x1 * x1);
    float y2 = wave_reduce_add(y0 * y0 + y1 * y1);
    float ca = 1.0f + 2.0f * CURV * xy + CURV * y2;
    float cb = 1.0f - CURV * x2;
    float dn = fmaxf(1.0f + 2.0f * CURV * xy + CURV * CURV * x2 * y2, MIN_NRM);
    float df0 = (ca * x0 + cb * y0) / dn;
    float df1 = (ca * x1 + cb * y1) / dn;
    float nrm = fmaxf(sqrtf(wave_reduce_add(df0 * df0 + df1 * df1)), MIN_NRM);
    float t = fminf(fmaxf(SQRTC * nrm, MIN_NRM), 1.0f - 1e-5f);
    float ls = atanhf(t) / t;
    float dt0 = ls * df0, dt1 = ls * df1;
    const v2f* avb = (const v2f*)(av + (size_t)r * 256);   // (r, h, dim)
#pragma unroll
    for (int hh = 0; hh < 4; ++hh) {
        v2f a = avb[hh * 32 + lane];
        float sc = wave_reduce_add(dt0 * a.x + dt1 * a.y);
        if (lane == 0) {
            sc = sc > 0.f ? sc : 0.2f * sc;               // leaky relu
            score[(size_t)e * 4 + hh] = sc;
            atomicMax(&m_enc[(size_t)d * 4 + hh], enc_f(sc));
        }
    }
}

// ---------- K3: exp(score - max) + segment denom ----------
__global__ void k_exp(const float* __restrict__ score, const unsigned* __restrict__ m_enc,
                      const int* __restrict__ dst, float* __restrict__ ex,
                      float* __restrict__ denom, int E) {
    int e = blockIdx.x * blockDim.x + threadIdx.x;
    if (e >= E) return;
    int d = dst[e];
#pragma unroll
    for (int hh = 0; hh < 4; ++hh) {
        float m = dec_f(m_enc[(size_t)d * 4 + hh]);
        float v = expf(score[(size_t)e * 4 + hh] - m);
        ex[(size_t)e * 4 + hh] = v;
        atomicAdd(&denom[(size_t)d * 4 + hh], v);
    }
}

// ---------- K4: WMMA message transform + exp-map + Einstein midpoint scatter ----------
// One wave = 16 sorted same-relation edges x one head (64 output dims).
// A (16x64 f32) = gathered h_t rows (zero-masked for pad slots, no EXEC predication);
// B (64x64) = rel_weight[r][head]. K=64 via 16 x V_WMMA_F32_16X16X4_F32 per 16-col tile.
__global__ void __launch_bounds__(256)
k_msg(const float* __restrict__ ht, const float* __restrict__ W,
      const int* __restrict__ src, const int* __restrict__ dst,
      const int* __restrict__ et, const int* __restrict__ sorted,
      const float* __restrict__ ex, const float* __restrict__ denom,
      float* __restrict__ num, float* __restrict__ den, int Tiles) {
    int wave = threadIdx.x >> 5, lane = threadIdx.x & 31;
    int gw = blockIdx.x * (blockDim.x >> 5) + wave;
    int tile = gw >> 2, head = gw & 3;
    if (tile >= Tiles) return;
    int q = lane & 15, half = lane >> 4;

    int sIdx = sorted[tile * 16 + q];          // lanes l and l+16 hold row q = l&15
    bool valid = sIdx >= 0;
    float vmask = valid ? 1.0f : 0.0f;
    int e = valid ? sIdx : 0;                  // pad rows alias edge 0 (safe memory)
    int srcv = src[e];
    int dstv = dst[e];
    int etv  = et[e];

    int fmin = wave_reduce_min(valid ? q : 1000);
    if (fmin == 1000) return;                  // fully-padded tile
    int r = __shfl(etv, fmin, 32);             // uniform relation for this tile

    const float* wb = W + (((size_t)r * 4 + head) << 12);  // 64x64 block
    const v2f* ab = (const v2f*)(ht + (size_t)srcv * 64);

    v8f cacc[4] = {};
#pragma unroll
    for (int k0 = 0; k0 < 64; k0 += 4) {
        int kk = k0 + half * 2;
        v2f a = ab[(k0 >> 1) + half];          // A layout: vgpr0=K(kk), vgpr1=K(kk+1)
        a.x *= vmask; a.y *= vmask;            // zero pad rows without touching EXEC
        const float* wk = wb + kk * 64 + q;
#pragma unroll
        for (int ot = 0; ot < 4; ++ot) {       // B layout mirrors A (row K across lanes)
            v2f b;
            b.x = wk[ot * 16];
            b.y = wk[64 + ot * 16];
            cacc[ot] = __builtin_amdgcn_wmma_f32_16x16x4_f32(
                false, a, false, b, (short)0, cacc[ot], false, false);
        }
    }

    // exp_map per edge row: rows (i + 8*half) live in cacc[*][i] across the 16-lane half
    float sc[8], lam[8];
#pragma unroll
    for (int i = 0; i < 8; ++i) {
        float p = cacc[0][i] * cacc[0][i] + cacc[1][i] * cacc[1][i]
                + cacc[2][i] * cacc[2][i] + cacc[3][i] * cacc[3][i];
        p = half_reduce_add(p);                // ||msg_t||^2 of row
        float nrm = fmaxf(sqrtf(p), MIN_NRM);
        float t = SQRTC * nrm;
        float s = tanhf(t) / t;                // exp-map scale
        sc[i] = s;
        float nsq_h = s * s * p;               // ||msg_h||^2
        lam[i] = 2.0f / (1.0f - CURV * nsq_h + EPSV);
    }

    // alpha for this lane's edge (row q), then redistribute to rows via shfl
    float exv = ex[(size_t)e * 4 + head] * vmask;
    float dnm = denom[(size_t)dstv * 4 + head];
    float alpha = exv / fmaxf(dnm, 1e-30f);

#pragma unroll
    for (int i = 0; i < 8; ++i) {
        int srcl = i + 8 * half;               // lane holding row (i + 8*half)
        float a_i = __shfl(alpha, srcl, 32);
        int   d_i = __shfl(dstv, srcl, 32);
        int   v_i = __shfl((int)valid, srcl, 32);
        if (!v_i) continue;
        float w_i = a_i * lam[i];
        float ws  = w_i * sc[i];
        float* np = num + ((size_t)d_i * 4 + head) * 64 + q;
        atomicAdd(&np[0],  ws * cacc[0][i]);
        atomicAdd(&np[16], ws * cacc[1][i]);
        atomicAdd(&np[32], ws * cacc[2][i]);
        atomicAdd(&np[48], ws * cacc[3][i]);
        if (q == 0) atomicAdd(&den[(size_t)d_i * 4 + head], w_i);  // once per row (half splits rows)
    }
}

// ---------- K5: midpoint -> project -> head-mean in tangent -> exp map ----------
__global__ void k_final(const float* __restrict__ num, const float* __restrict__ den,
                        float* __restrict__ out, int N) {
    int wave = threadIdx.x >> 5, lane = threadIdx.x & 31;
    int n = blockIdx.x * (blockDim.x >> 5) + wave;
    if (n >= N) return;
    const float maxnorm = (1.0f - 1e-5f) / SQRTC;
    float a0 = 0.f, a1 = 0.f;
#pragma unroll
    for (int hh = 0; hh < 4; ++hh) {
        v2f nv = ((const v2f*)(num + ((size_t)n * 4 + hh) * 64))[lane];
        float dv = den[(size_t)n * 4 + hh] + EPSV;
        float v0 = nv.x / dv, v1 = nv.y / dv;
        float nrm = fmaxf(sqrtf(wave_reduce_add(v0 * v0 + v1 * v1)), MIN_NRM);
        if (nrm > maxnorm) { float f = maxnorm / nrm; v0 *= f; v1 *= f; nrm = maxnorm; }
        float t = fminf(fmaxf(SQRTC * nrm, MIN_NRM), 1.0f - 1e-5f);
        float ls = atanhf(t) / t;
        a0 += 0.25f * ls * v0; a1 += 0.25f * ls * v1;
    }
    float nrm = fmaxf(sqrtf(wave_reduce_add(a0 * a0 + a1 * a1)), MIN_NRM);
    float t = SQRTC * nrm;
    float es = tanhf(t) / t;
    v2f o; o.x = es * a0; o.y = es * a1;
    ((v2f*)(out + (size_t)n * 64))[lane] = o;
}

extern "C" void kernel_launch(void* const* d_in, const int* in_sizes, int n_in,
                              void* d_out, int out_size, void* d_ws, size_t ws_size,
                              hipStream_t stream) {
    const float* h  = (const float*)d_in[0];   // (N, 64) hyperbolic features
    const float* W  = (const float*)d_in[1];   // (8, 4, 64, 64)
    const float* av = (const float*)d_in[2];   // (8, 4, 64)
    const int* src  = (const int*)d_in[4];
    const int* dst  = (const int*)d_in[5];
    const int* et   = (const int*)d_in[6];
    int N = in_sizes[0] / 64;
    int E = in_sizes[4];

    char* p = (char*)d_ws;
    auto alloc = [&](size_t bytes) { char* q = p; p += (bytes + 255) & ~(size_t)255; return q; };
    float*    ht     = (float*)alloc((size_t)N * 64 * 4);
    float*    score  = (float*)alloc((size_t)E * 4 * 4);
    float*    exv    = (float*)alloc((size_t)E * 4 * 4);
    unsigned* m_enc  = (unsigned*)alloc((size_t)N * 4 * 4);
    float*    denom  = (float*)alloc((size_t)N * 4 * 4);
    float*    num    = (float*)alloc((size_t)N * 256 * 4);
    float*    den    = (float*)alloc((size_t)N * 4 * 4);
    unsigned* cnt    = (unsigned*)alloc(8 * 4);
    unsigned* cursor = (unsigned*)alloc(8 * 4);
    int Lsorted = (int)(((size_t)E + 135) / 16) * 16;  // covers 8 relations padded to 16
    int* sorted = (int*)alloc((size_t)Lsorted * 4);

    k_init   <<<2048, 256, 0, stream>>>(m_enc, denom, num, den, cnt, sorted, N, Lsorted);
    k_logmap <<<(N + 7) / 8, 256, 0, stream>>>(h, ht, N);
    k_hist   <<<(E + 255) / 256, 256, 0, stream>>>(et, cnt, E);
    k_offsets<<<1, 1, 0, stream>>>(cnt, cursor);
    k_scatter<<<(E + 255) / 256, 256, 0, stream>>>(et, cursor, sorted, E);
    k_score  <<<(E + 7) / 8, 256, 0, stream>>>(h, av, src, dst, et, score, m_enc, E);
    k_exp    <<<(E + 255) / 256, 256, 0, stream>>>(score, m_enc, dst, exv, denom, E);
    int Tiles = Lsorted / 16;
    int waves = Tiles * 4;
    k_msg    <<<(waves + 7) / 8, 256, 0, stream>>>(ht, W, src, dst, et, sorted,
                                                   exv, denom, num, den, Tiles);
    k_final  <<<(N + 7) / 8, 256, 0, stream>>>(num, den, (float*)d_out, N);
}